// EdgeConvRegressor_55241869361463
// MI455X (gfx1250) — compile-verified
//
#include <hip/hip_runtime.h>
#include <hip/hip_bf16.h>
#include <math.h>

// Problem constants (match reference)
#define BB 32
#define NN 2048
#define IND 16
#define HID 64
#define KNN 8

typedef float v2f __attribute__((ext_vector_type(2)));
typedef float v8f __attribute__((ext_vector_type(8)));

// ---------------------------------------------------------------------------
// fp32 WMMA 16x16x4:  D(16x16) = A(16x4) * B(4x16) + C
// A layout: lanes 0-15 -> M=lane, v0=K0 v1=K1 ; lanes 16-31 -> M=lane-16, v0=K2 v1=K3
// B layout (mirror):  lanes 0-15 -> N=lane, v0=K0 v1=K1 ; lanes 16-31 -> N=lane-16, v0=K2 v1=K3
// C layout: lane<16: rows 0..7 col=lane ; lane>=16: rows 8..15 col=lane-16
// ---------------------------------------------------------------------------
__device__ __forceinline__ v8f wmma4(v2f a, v2f b, v8f c) {
  return __builtin_amdgcn_wmma_f32_16x16x4_f32(false, a, false, b, (short)0, c,
                                               false, false);
}

// A-tile loader from row-major [rows x ld] matrix, rows r0..r0+15, K-step s (k0=4s)
__device__ __forceinline__ v2f ldA(const float* __restrict__ base, int ld,
                                   int r0, int s, int lane) {
  int row = r0 + (lane & 15);
  int k = 4 * s + ((lane >> 4) << 1);
  const float* p = base + (size_t)row * ld + k;
  v2f v; v.x = p[0]; v.y = p[1];
  return v;
}

// B-tile loader from row-major weight W[K x ldw]: B[k][n] = W[k][n0+n]
__device__ __forceinline__ v2f ldBw(const float* __restrict__ W, int ldw,
                                    int n0, int s, int lane) {
  int n = n0 + (lane & 15);
  int k = 4 * s + ((lane >> 4) << 1);
  v2f v; v.x = W[(size_t)k * ldw + n]; v.y = W[(size_t)(k + 1) * ldw + n];
  return v;
}

// ---------------------------------------------------------------------------
// Prep: Wd = W_top - W_bot for both edge convs (split of the concat-GEMM)
// ---------------------------------------------------------------------------
__global__ void prep_kernel(const float* __restrict__ Wc1a,
                            const float* __restrict__ Wc2a,
                            float* __restrict__ Wd1, float* __restrict__ Wd2) {
  int t = blockIdx.x * blockDim.x + threadIdx.x;
  if (t < HID * HID) {
    Wd1[t] = Wc1a[t] - Wc1a[HID * HID + t];
    Wd2[t] = Wc2a[t] - Wc2a[HID * HID + t];
  }
}

// ---------------------------------------------------------------------------
// Input MLP: h = relu(relu(x@Wi1+bi1)@Wi2+bi2), plus n2 = ||h||^2 per point.
// One wave per 16 points.
// ---------------------------------------------------------------------------
__global__ __launch_bounds__(32) void mlp_in_kernel(
    const float* __restrict__ x, const float* __restrict__ Wi1,
    const float* __restrict__ bi1, const float* __restrict__ Wi2,
    const float* __restrict__ bi2, float* __restrict__ h0,
    float* __restrict__ n2) {
  const int lane = threadIdx.x;
  const int p0 = blockIdx.x << 4;
  const int half = lane >> 4, col16 = lane & 15;

  __shared__ float T[16][HID + 1];
  __shared__ float S[16][16];

  // stage 1: x(16x16) @ Wi1(16x64)
  v2f a1[4];
#pragma unroll
  for (int s = 0; s < 4; ++s) a1[s] = ldA(x, IND, p0, s, lane);
#pragma unroll
  for (int n0 = 0; n0 < HID; n0 += 16) {
    v8f c = {};
#pragma unroll
    for (int s = 0; s < 4; ++s) c = wmma4(a1[s], ldBw(Wi1, HID, n0, s, lane), c);
    float bb = bi1[n0 + col16];
#pragma unroll
    for (int v = 0; v < 8; ++v)
      T[half * 8 + v][n0 + col16] = fmaxf(c[v] + bb, 0.0f);
  }
  __syncthreads();

  // stage 2: T(16x64) @ Wi2(64x64), write h and accumulate ||h||^2
  v2f a2[16];
#pragma unroll
  for (int s = 0; s < 16; ++s) {
    int k = 4 * s + 2 * half;
    v2f t; t.x = T[col16][k]; t.y = T[col16][k + 1];
    a2[s] = t;
  }
  float acc[8];
#pragma unroll
  for (int v = 0; v < 8; ++v) acc[v] = 0.0f;
#pragma unroll
  for (int n0 = 0; n0 < HID; n0 += 16) {
    v8f c = {};
#pragma unroll
    for (int s = 0; s < 16; ++s) c = wmma4(a2[s], ldBw(Wi2, HID, n0, s, lane), c);
    float bb = bi2[n0 + col16];
#pragma unroll
    for (int v = 0; v < 8; ++v) {
      float val = fmaxf(c[v] + bb, 0.0f);
      h0[(size_t)(p0 + half * 8 + v) * HID + n0 + col16] = val;
      acc[v] += val * val;
    }
  }
#pragma unroll
  for (int v = 0; v < 8; ++v) S[half * 8 + v][col16] = acc[v];
  __syncthreads();
  if (lane < 16) {
    float s = 0.0f;
    for (int u = 0; u < 16; ++u) s += S[lane][u];
    n2[p0 + lane] = s;
  }
}

// ---------------------------------------------------------------------------
// Fused kNN: per wave, one batch + 16-row i-tile. Streams j-tiles through
// fp32 WMMA (dot products), converts to distances, keeps running top-8 per
// row (2 lanes per row, merged at the end). Never materializes NxN.
// ---------------------------------------------------------------------------
__global__ __launch_bounds__(32) void knn_kernel(const float* __restrict__ H,
                                                 const float* __restrict__ n2,
                                                 int* __restrict__ idxOut) {
  const int lane = threadIdx.x;
  const int b = blockIdx.x >> 7;          // 128 i-tiles per batch
  const int i0 = (blockIdx.x & 127) << 4;
  const float* Hb = H + (size_t)b * NN * HID;
  const float* n2b = n2 + (size_t)b * NN;
  const int half = lane >> 4, col16 = lane & 15;

  v2f areg[16];
#pragma unroll
  for (int s = 0; s < 16; ++s) areg[s] = ldA(Hb, HID, i0, s, lane);

  float n2i[8];
#pragma unroll
  for (int v = 0; v < 8; ++v) n2i[v] = n2b[i0 + half * 8 + v];

  // per-lane running top-8 (sorted descending: bd[0] = current max)
  float bd[8]; int bj[8];
#pragma unroll
  for (int v = 0; v < 8; ++v) { bd[v] = 3.0e38f; bj[v] = 0; }

  __shared__ float Dt[16][17];

  for (int j0 = 0; j0 < NN; j0 += 16) {
    v8f c = {};
#pragma unroll
    for (int s = 0; s < 16; ++s) c = wmma4(areg[s], ldA(Hb, HID, j0, s, lane), c);

    int col = j0 + col16;
    float n2j = n2b[col];
#pragma unroll
    for (int v = 0; v < 8; ++v) {
      int rowg = i0 + half * 8 + v;
      float d = n2i[v] + n2j - 2.0f * c[v];
      if (rowg == col) d = 1.0e30f;   // exclude self (ref adds 1e30*eye)
      Dt[half * 8 + v][col16] = d;
    }
    __syncthreads();

    // lane (and lane+16) own row col16; scan 8 columns each
    const int cb = half * 8;
#pragma unroll
    for (int t = 0; t < 8; ++t) {
      float d = Dt[col16][cb + t];
      if (d < bd[0]) {
        bd[0] = d; bj[0] = j0 + cb + t;
#pragma unroll
        for (int u = 0; u < 7; ++u)
          if (bd[u] < bd[u + 1]) {
            float td = bd[u]; bd[u] = bd[u + 1]; bd[u + 1] = td;
            int tj = bj[u]; bj[u] = bj[u + 1]; bj[u + 1] = tj;
          }
      }
    }
    __syncthreads();
  }

  // merge the two half-lane top-8 lists per row, emit 8 smallest indices
  __shared__ float Md[16][16];
  __shared__ int Mj[16][16];
#pragma unroll
  for (int t = 0; t < 8; ++t) {
    Md[col16][half * 8 + t] = bd[t];
    Mj[col16][half * 8 + t] = bj[t];
  }
  __syncthreads();
  if (lane < 16) {
    int* op = idxOut + ((size_t)b * NN + i0 + lane) * KNN;
    for (int t = 0; t < KNN; ++t) {
      float best = 3.0e38f; int bi = 0;
      for (int u = 0; u < 16; ++u) {
        float d = Md[lane][u];
        if (d < best) { best = d; bi = u; }
      }
      op[t] = Mj[lane][bi];
      Md[lane][bi] = 3.0e38f;
    }
  }
}

// ---------------------------------------------------------------------------
// P/Q precompute for one edge conv: P = h@(W1top-W1bot)+b1, Q = h@W1bot.
// One wave per 16 points.
// ---------------------------------------------------------------------------
__global__ __launch_bounds__(32) void pq_kernel(
    const float* __restrict__ h, const float* __restrict__ Wd,
    const float* __restrict__ W1bot, const float* __restrict__ b1,
    float* __restrict__ P, float* __restrict__ Q) {
  const int lane = threadIdx.x;
  const int p0 = blockIdx.x << 4;
  const int half = lane >> 4, col16 = lane & 15;

  v2f a[16];
#pragma unroll
  for (int s = 0; s < 16; ++s) a[s] = ldA(h, HID, p0, s, lane);

#pragma unroll
  for (int n0 = 0; n0 < HID; n0 += 16) {
    v8f c = {};
#pragma unroll
    for (int s = 0; s < 16; ++s) c = wmma4(a[s], ldBw(Wd, HID, n0, s, lane), c);
    float bb = b1[n0 + col16];
#pragma unroll
    for (int v = 0; v < 8; ++v)
      P[(size_t)(p0 + half * 8 + v) * HID + n0 + col16] = c[v] + bb;
  }
#pragma unroll
  for (int n0 = 0; n0 < HID; n0 += 16) {
    v8f c = {};
#pragma unroll
    for (int s = 0; s < 16; ++s) c = wmma4(a[s], ldBw(W1bot, HID, n0, s, lane), c);
#pragma unroll
    for (int v = 0; v < 8; ++v)
      Q[(size_t)(p0 + half * 8 + v) * HID + n0 + col16] = c[v];
  }
}

// ---------------------------------------------------------------------------
// Edge kernel: one wave per 2 points (= 16 edges). Rows 0..7 are point p0's
// neighbors, rows 8..15 are p0+1's. T = relu(P_i+Q_j); out = mean relu(T@W2+b2).
// WMMA C-layout puts each point's 8 edge results in 8 VGPRs of one lane, so
// the neighbor-mean is an in-lane reduction. Optionally emits ||h||^2.
// ---------------------------------------------------------------------------
__global__ __launch_bounds__(32) void edge_kernel(
    const float* __restrict__ P, const float* __restrict__ Q,
    const int* __restrict__ idx, const float* __restrict__ W2,
    const float* __restrict__ b2, float* __restrict__ hout,
    float* __restrict__ n2out) {
  const int lane = threadIdx.x;
  const int p0 = blockIdx.x * 2;
  const int half = lane >> 4, col16 = lane & 15;

  __shared__ float S[2][16];

  const int row = col16;                       // edge row 0..15
  const int i = p0 + (row >> 3);               // center point (global)
  const int j = (i & ~(NN - 1)) + idx[(size_t)i * KNN + (row & 7)];
  const float* Pi = P + (size_t)i * HID;
  const float* Qj = Q + (size_t)j * HID;

  v2f a[16];
#pragma unroll
  for (int s = 0; s < 16; ++s) {
    int k = 4 * s + 2 * half;
    v2f t;
    t.x = fmaxf(Pi[k] + Qj[k], 0.0f);
    t.y = fmaxf(Pi[k + 1] + Qj[k + 1], 0.0f);
    a[s] = t;
  }

  float acc = 0.0f;
#pragma unroll
  for (int n0 = 0; n0 < HID; n0 += 16) {
    v8f c = {};
#pragma unroll
    for (int s = 0; s < 16; ++s) c = wmma4(a[s], ldBw(W2, HID, n0, s, lane), c);
    float bb = b2[n0 + col16];
    float sum = 0.0f;
#pragma unroll
    for (int v = 0; v < 8; ++v) sum += fmaxf(c[v] + bb, 0.0f);
    float val = sum * 0.125f;  // mean over 8 neighbors
    hout[(size_t)(p0 + half) * HID + n0 + col16] = val;
    acc += val * val;
  }

  if (n2out != nullptr) {
    S[half][col16] = acc;
    __syncthreads();
    if (lane < 2) {
      float s = 0.0f;
      for (int u = 0; u < 16; ++u) s += S[lane][u];
      n2out[p0 + lane] = s;
    }
  }
}

// ---------------------------------------------------------------------------
// Head: y = relu(h@Wo1+bo1)@Wo2+bo2 ; out = softplus(y)*mask.
// One wave per 16 points.
// ---------------------------------------------------------------------------
__global__ __launch_bounds__(32) void head_kernel(
    const float* __restrict__ h2, const float* __restrict__ Wo1,
    const float* __restrict__ bo1, const float* __restrict__ Wo2,
    const float* __restrict__ bo2, const float* __restrict__ x,
    float* __restrict__ out) {
  const int lane = threadIdx.x;
  const int p0 = blockIdx.x << 4;
  const int half = lane >> 4, col16 = lane & 15;

  __shared__ float T[16][HID + 1];
  __shared__ float S2[2][16];

  v2f a[16];
#pragma unroll
  for (int s = 0; s < 16; ++s) a[s] = ldA(h2, HID, p0, s, lane);
#pragma unroll
  for (int n0 = 0; n0 < HID; n0 += 16) {
    v8f c = {};
#pragma unroll
    for (int s = 0; s < 16; ++s) c = wmma4(a[s], ldBw(Wo1, HID, n0, s, lane), c);
    float bb = bo1[n0 + col16];
#pragma unroll
    for (int v = 0; v < 8; ++v)
      T[half * 8 + v][n0 + col16] = fmaxf(c[v] + bb, 0.0f);
  }
  __syncthreads();

  // final 64->1 projection: lane handles row col16, half of the 64 columns
  float part = 0.0f;
#pragma unroll
  for (int u = 0; u < 32; ++u) {
    int cc = half * 32 + u;
    part += T[col16][cc] * Wo2[cc];
  }
  S2[half][col16] = part;
  __syncthreads();
  if (lane < 16) {
    float y = S2[0][lane] + S2[1][lane] + bo2[0];
    float sp = fmaxf(y, 0.0f) + log1pf(expf(-fabsf(y)));
    int p = p0 + lane;
    float mask = (x[(size_t)p * IND + (IND - 1)] > 0.5f) ? 1.0f : 0.0f;
    out[p] = sp * mask;
  }
}

// ---------------------------------------------------------------------------
extern "C" void kernel_launch(void* const* d_in, const int* in_sizes, int n_in,
                              void* d_out, int out_size, void* d_ws,
                              size_t ws_size, hipStream_t stream) {
  const float* x    = (const float*)d_in[0];
  const float* Wi1  = (const float*)d_in[1];
  const float* bi1  = (const float*)d_in[2];
  const float* Wi2  = (const float*)d_in[3];
  const float* bi2  = (const float*)d_in[4];
  const float* Wc1a = (const float*)d_in[5];
  const float* bc1a = (const float*)d_in[6];
  const float* Wc1b = (const float*)d_in[7];
  const float* bc1b = (const float*)d_in[8];
  const float* Wc2a = (const float*)d_in[9];
  const float* bc2a = (const float*)d_in[10];
  const float* Wc2b = (const float*)d_in[11];
  const float* Wo1  = (const float*)d_in[13];
  const float* bo1  = (const float*)d_in[14];
  const float* Wo2  = (const float*)d_in[15];
  const float* bo2  = (const float*)d_in[16];
  // d_in[12] = bc2b, d_in[17] = k (=8, baked in)
  const float* bc2b = (const float*)d_in[12];

  float* ws = (float*)d_ws;
  const size_t nPts = (size_t)BB * NN;      // 65536
  const size_t szH  = nPts * HID;           // 4,194,304 floats
  float* h0  = ws;
  float* h1  = ws + szH;
  float* Pb  = ws + 2 * szH;
  float* Qb  = ws + 3 * szH;
  float* n2a = ws + 4 * szH;
  float* n2b = n2a + nPts;
  float* Wd1 = n2b + nPts;
  float* Wd2 = Wd1 + HID * HID;
  int* idx0  = (int*)(Wd2 + HID * HID);
  int* idx1  = idx0 + nPts * KNN;
  float* h2  = h0;  // h0 dead after pq of conv1; reuse for conv2 output

  const int tiles16 = (int)(nPts / 16);   // 4096
  const int pairs   = (int)(nPts / 2);    // 32768

  prep_kernel<<<(HID * HID + 255) / 256, 256, 0, stream>>>(Wc1a, Wc2a, Wd1, Wd2);

  mlp_in_kernel<<<tiles16, 32, 0, stream>>>(x, Wi1, bi1, Wi2, bi2, h0, n2a);

  // conv 1
  knn_kernel<<<BB * (NN / 16), 32, 0, stream>>>(h0, n2a, idx0);
  pq_kernel<<<tiles16, 32, 0, stream>>>(h0, Wd1, Wc1a + HID * HID, bc1a, Pb, Qb);
  edge_kernel<<<pairs, 32, 0, stream>>>(Pb, Qb, idx0, Wc1b, bc1b, h1, n2b);

  // conv 2
  knn_kernel<<<BB * (NN / 16), 32, 0, stream>>>(h1, n2b, idx1);
  pq_kernel<<<tiles16, 32, 0, stream>>>(h1, Wd2, Wc2a + HID * HID, bc2a, Pb, Qb);
  edge_kernel<<<pairs, 32, 0, stream>>>(Pb, Qb, idx1, Wc2b, bc2b, h2, nullptr);

  head_kernel<<<tiles16, 32, 0, stream>>>(h2, Wo1, bo1, Wo2, bo2, x,
                                          (float*)d_out);
}